// FrankeWesterhoff_76785425318195
// MI455X (gfx1250) — compile-verified
//
#include <hip/hip_runtime.h>
#include <cstdint>

// Franke-Westerhoff scan: B=16384 independent series, sequential over T-1=2047
// steps. One lane per series, one wave32 per block (512 blocks -> all WGPs).
// es tiles (32 series x 32 timesteps, rows 8KB apart) are DMA'd by the Tensor
// Data Mover into LDS with 1-DWORD-per-32 padding (row stride 33 dwords ->
// bank-conflict-free ds_load_b32), double-buffered on TENSORcnt.

namespace {

constexpr int kB     = 16384;
constexpr int kT     = 2048;
constexpr int kOutW  = kT - 1;               // 2047 outputs per series
constexpr int kTile  = 32;                   // timesteps per TDM tile
constexpr int kRows  = 32;                   // series per block == one wave32
constexpr int kStride = kTile + 1;           // padded LDS row stride (dwords)
constexpr int kBufDw  = kRows * kStride;     // dwords per buffer (1056)
constexpr int kNTiles = (kOutW + kTile - 1) / kTile;  // 64

// model constants
constexpr float kMuPhi  = 0.01f * 0.12f;     // MU*PHI
constexpr float kMuChi  = 0.01f * 1.5f;      // MU*CHI
constexpr float kSigF2  = 0.758f * 0.758f;   // SIGMAF^2
constexpr float kMu     = 0.01f;
constexpr float kNegL2E = -1.44269504088896f; // -log2(e), BETA = 1

typedef unsigned int v4u __attribute__((ext_vector_type(4)));
typedef int          v8i __attribute__((ext_vector_type(8)));
typedef int          v4i __attribute__((ext_vector_type(4)));

// Issue one TDM 2D tile load: kRows rows x kTile fp32 columns from global
// (row pitch = kT elements) into LDS at lds_byte_off, padding +1 dword after
// every 32 dwords. valid_w (<= kTile) = tensor_dim0 so OOB columns zero-fill.
__device__ __forceinline__ void tdm_load_tile(unsigned int lds_byte_off,
                                              const float* gsrc, int valid_w) {
  const unsigned long long ga = (unsigned long long)(uintptr_t)gsrc;

  v4u g0;
  g0[0] = 1u;                                        // count=1, user descriptor
  g0[1] = lds_byte_off;                              // lds_addr
  g0[2] = (unsigned int)(ga & 0xffffffffull);        // global_addr[31:0]
  g0[3] = (unsigned int)((ga >> 32) & 0x1ffffffull)  // global_addr[56:32]
          | (2u << 30);                              // type=2 ("image")

  v8i g1;
  // data_size=2 (4B) | pad_enable | pad_interval=4 (32 dwords) | pad_amount=0 (1 dword)
  g1[0] = (int)((2u << 16) | (1u << 20) | (4u << 22));
  g1[1] = (int)(((unsigned int)valid_w & 0xffffu) << 16);  // tensor_dim0[15:0]
  g1[2] = (int)((unsigned int)kRows << 16);                // tensor_dim1[15:0]
  g1[3] = (int)((unsigned int)kTile << 16);                // tile_dim0
  g1[4] = kRows;                                           // tile_dim1 (tile_dim2=0)
  g1[5] = kT;                                              // tensor_dim0_stride[31:0]
  g1[6] = 0;
  g1[7] = 0;

  const v4i gz4 = {0, 0, 0, 0};                      // 2D tensor: groups 2/3 zero
  const v8i gz8 = {0, 0, 0, 0, 0, 0, 0, 0};          // trailing group (clang-23 6-arg form)
  __builtin_amdgcn_tensor_load_to_lds(g0, g1, gz4, gz4, gz8, 0);
}

__global__ __launch_bounds__(kRows) void fw_scan_kernel(
    const float* __restrict__ pars, const float* __restrict__ es,
    float* __restrict__ out) {
  __shared__ float lds[2 * kBufDw];

  const int lane = threadIdx.x;                 // wave32: lane == tid
  const int row0 = blockIdx.x * kRows;
  const int b    = row0 + lane;

  const float4 pv = reinterpret_cast<const float4*>(pars)[b];
  const float alpha0 = pv.x, alphan = pv.y, alphap = pv.z, sigmac = pv.w;
  const float sc2 = sigmac * sigmac;

  const float* es_base = es + (size_t)row0 * kT;
  const unsigned int lds_base = (unsigned int)(uintptr_t)(&lds[0]);

  float p1 = 0.0f, p2 = 0.0f, a = alpha0;

  // prefetch tile 0 into buffer 0
  tdm_load_tile(lds_base, es_base, kTile);

  unsigned int phase = 0;
  int t0 = 0;
  for (int k = 0; k < kNTiles; ++k) {
    const int next_t0 = t0 + kTile;
    if (k + 1 < kNTiles) {
      int wn = kOutW - next_t0;
      if (wn > kTile) wn = kTile;
      tdm_load_tile(lds_base + (phase ^ 1u) * (kBufDw * 4u),
                    es_base + next_t0, wn);
      __builtin_amdgcn_s_wait_tensorcnt(1);     // tile k landed
    } else {
      __builtin_amdgcn_s_wait_tensorcnt(0);     // last tile landed
    }

    const float* ebuf = &lds[phase * kBufDw + lane * kStride];
    float* orow = out + (size_t)b * kOutW + t0;

#pragma unroll
    for (int j = 0; j < kTile; ++j) {
      const float e  = ebuf[j];                               // ds_load_b32, no conflicts
      const float ex = __builtin_amdgcn_exp2f(a * kNegL2E);   // exp(-a)
      const float nf = __builtin_amdgcn_rcpf(1.0f + ex);      // sigmoid(a)
      const float g  = 1.0f - nf;
      const float c  = kMuChi * g;
      const float mean = p1 * (1.0f + c - kMuPhi * nf) - c * p2;
      const float var  = nf * nf * kSigF2 + g * g * sc2;
      const float sd   = kMu * __builtin_amdgcn_sqrtf(var);
      const float pn   = __builtin_fmaf(sd, e, mean);
      const float an   = __builtin_fmaf(
          alphap, p1 * p1, __builtin_fmaf(alphan, 2.0f * nf - 1.0f, alpha0));
      if (t0 + j < kOutW) orow[j] = pn;         // uniform predicate (tail tile)
      p2 = p1;
      p1 = pn;
      a  = an;
    }

    phase ^= 1u;
    t0 = next_t0;
  }
}

}  // namespace

extern "C" void kernel_launch(void* const* d_in, const int* in_sizes, int n_in,
                              void* d_out, int out_size, void* d_ws,
                              size_t ws_size, hipStream_t stream) {
  const float* pars = (const float*)d_in[0];   // (B,4) fp32
  const float* es   = (const float*)d_in[1];   // (B,T) fp32
  float* out        = (float*)d_out;           // (B,T-1) fp32
  (void)in_sizes; (void)n_in; (void)out_size; (void)d_ws; (void)ws_size;

  dim3 grid(kB / kRows);   // 512 single-wave workgroups
  dim3 block(kRows);       // 32 threads = one wave32
  hipLaunchKernelGGL(fw_scan_kernel, grid, block, 0, stream, pars, es, out);
}